// TorchSMoE_SMoE_50337016709598
// MI455X (gfx1250) — compile-verified
//
#include <hip/hip_runtime.h>

typedef __attribute__((ext_vector_type(2))) float v2f;
typedef __attribute__((ext_vector_type(8))) float v8f;

#define NKER 4
#define BLK_PER_WAVE 4     // 4 blocks -> 16 WMMA columns (4 blocks x 4 kernels)
#define WAVES_PER_WG 8

// DPP quad_perm cross-lane adds (kernel index = lane&3, so the softmax
// reduction lives entirely inside 4-lane quads -> pure-VALU butterflies).
__device__ __forceinline__ float quad_xor1(float v) {
    int i = __builtin_bit_cast(int, v);
    i = __builtin_amdgcn_update_dpp(0, i, 0xB1, 0xF, 0xF, true); // [1,0,3,2]
    return __builtin_bit_cast(float, i);
}
__device__ __forceinline__ float quad_xor2(float v) {
    int i = __builtin_bit_cast(int, v);
    i = __builtin_amdgcn_update_dpp(0, i, 0x4E, 0xF, 0xF, true); // [2,3,0,1]
    return __builtin_bit_cast(float, i);
}

// One wave evaluates 4 blocks:
//   Q(64x16) = P(64x8) x C(8x16) via 8x V_WMMA_F32_16X16X4_F32
// P = fixed quadratic basis [X^2, XY, Y^2, X, Y, 1, 0, 0] over the 8x8 domain.
// C = per-(block,kernel) coefficients of q, pre-scaled by -0.5*log2(e) so the
// accumulator is directly the exp2 argument.
__global__ __launch_bounds__(256) void smoe_wmma_kernel(const float* __restrict__ x,
                                                        float* __restrict__ out,
                                                        int n_groups, int n_blocks) {
    const int lane = threadIdx.x & 31;
    const int wave = threadIdx.x >> 5;
    const int group = blockIdx.x * WAVES_PER_WG + wave;
    if (group >= n_groups) return;                 // wave-uniform

    __shared__ __align__(16) float smem[WAVES_PER_WG * 256];
    float* sl = smem + wave * 256;

    const int  ncol = lane & 15;                   // WMMA column this lane owns
    const bool hi   = lane >= 16;
    const int  b    = ncol >> 2;                   // block within group
    const int  k    = ncol & 3;                    // kernel index
    int blk = group * BLK_PER_WAVE + b;
    if (blk >= n_blocks) blk = n_blocks - 1;       // clamp (loads only)
    const long base = (long)blk * (7 * NKER);

    // Parameter loads: lanes 16-31 mirror 0-15 -> L0 broadcast, few cache lines.
    const float cx  = x[base + k];
    const float cy  = x[base + NKER + k];
    const float nue = x[base + 2 * NKER + k];
    const float A00 = x[base + 3 * NKER + 4 * k + 0];
    const float A10 = x[base + 3 * NKER + 4 * k + 2];   // tril: A01 = 0
    const float A11 = x[base + 3 * NKER + 4 * k + 3];

    // q = (A00*u + A10*v)^2 + (A11*v)^2, u=X-cx, v=Y-cy -> quadratic in (X,Y).
    const float s  = -0.72134752044448170368f;     // -0.5 * log2(e)
    const float al = s * A00 * A00;
    const float be = s * 2.0f * A00 * A10;
    const float ga = s * (A10 * A10 + A11 * A11);
    const float de = -2.0f * al * cx - be * cy;
    const float ep = -be * cx - 2.0f * ga * cy;
    const float ze = al * cx * cx + be * cx * cy + ga * cy * cy;

    // B operand (4x16, K rows): VGPR0={K0|K2}, VGPR1={K1|K3} across lane halves.
    v2f B0, B1;
    B0.x = hi ? ga : al;   B0.y = hi ? de : be;    // rows {a,b,g,d}
    B1.x = hi ? 0.f : ep;  B1.y = hi ? 0.f : ze;   // rows {e,z,0,0}

    const float inv7 = 1.0f / 7.0f;
    v8f acc[4];
#pragma unroll
    for (int t = 0; t < 4; ++t) {
        const int p  = 16 * t + (lane & 15);       // A operand: lane = M row
        const float X = (float)(p >> 3) * inv7;
        const float Y = (float)(p & 7) * inv7;
        const float xx = X * X, xy = X * Y, yy = Y * Y;
        v2f A0, A1;
        A0.x = hi ? yy  : xx;                      // cols {X^2, XY, Y^2, X}
        A0.y = hi ? X   : xy;
        A1.x = hi ? 0.f : Y;                       // cols {Y, 1, 0, 0}
        A1.y = hi ? 0.f : 1.f;
        v8f c = {0.f, 0.f, 0.f, 0.f, 0.f, 0.f, 0.f, 0.f};
        c = __builtin_amdgcn_wmma_f32_16x16x4_f32(false, A0, false, B0, (short)0, c, false, false);
        c = __builtin_amdgcn_wmma_f32_16x16x4_f32(false, A1, false, B1, (short)0, c, false, false);
        acc[t] = c;
    }

    // Epilogue: e = exp2(acc) = exp(-q/2); softmax over the 4-lane kernel quad
    // via DPP quad_perm butterflies (no LDS traffic, no DScnt stalls).
#pragma unroll
    for (int t = 0; t < 4; ++t) {
        float r[8];
#pragma unroll
        for (int v = 0; v < 8; ++v) {
            float e   = __builtin_amdgcn_exp2f(acc[t][v]);
            float num = e * nue;
            e   += quad_xor1(e);
            num += quad_xor1(num);
            e   += quad_xor2(e);
            num += quad_xor2(num);
            float den = fmaxf(e, 1e-8f);
            float rr  = num * __builtin_amdgcn_rcpf(den);
            r[v] = fminf(fmaxf(rr, 0.f), 1.f);
        }
        if (k == 0) {                              // one writer per quad
            const int pbase = 16 * t + (hi ? 8 : 0);   // C/D: lanes>=16 -> M+8
            float4* dst = (float4*)(sl + b * 64 + pbase);
            dst[0] = make_float4(r[0], r[1], r[2], r[3]);
            dst[1] = make_float4(r[4], r[5], r[6], r[7]);
        }
    }

    // Same-wave DS ordering before cross-lane readback (LDS is in-order per
    // wave; the asm wait + memory clobber also blocks compiler reordering).
    asm volatile("s_wait_dscnt 0" ::: "memory");

    // Coalesced 128-bit global stores: 4 blocks x 64 floats = 64 float4.
    const float4* s4   = (const float4*)sl;
    float4*       outv = (float4*)out;
    const long lim = (long)n_blocks * 16;          // out size in float4 units
    const long i0  = (long)group * 64 + lane;
    const long i1  = i0 + 32;
    if (i0 < lim) outv[i0] = s4[lane];
    if (i1 < lim) outv[i1] = s4[lane + 32];
}

extern "C" void kernel_launch(void* const* d_in, const int* in_sizes, int n_in,
                              void* d_out, int out_size, void* d_ws, size_t ws_size,
                              hipStream_t stream) {
    const float* x = (const float*)d_in[0];
    float* out = (float*)d_out;
    const int n_blocks = in_sizes[0] / (7 * NKER);           // 1048576
    const int n_groups = (n_blocks + BLK_PER_WAVE - 1) / BLK_PER_WAVE;
    const int n_wgs    = (n_groups + WAVES_PER_WG - 1) / WAVES_PER_WG;
    smoe_wmma_kernel<<<n_wgs, WAVES_PER_WG * 32, 0, stream>>>(x, out, n_groups, n_blocks);
}